// Conv1D_71305047048638
// MI455X (gfx1250) — compile-verified
//
#include <hip/hip_runtime.h>
#include <cstdint>
#include <cstddef>

// Causal depthwise temporal conv, K=4, f32.
// x: [4][4096][4096], w: [4][4096], b: [4096] -> out: [4][4096][4096]
// out[t,d] = b[d] + w[0,d]*x[t-3,d] + w[1,d]*x[t-2,d] + w[2,d]*x[t-1,d] + w[3,d]*x[t,d]
//
// Memory-bound (~1 FLOP/byte). Strategy: stream x via the CDNA5 async
// global->LDS DMA path (ASYNCcnt), depth-7 LDS ring buffer, register
// sliding window so each x byte is fetched once, 128-bit coalesced
// loads/stores, non-temporal stores.

typedef __attribute__((ext_vector_type(4))) float v4f;

#define D_TOTAL 4096
#define T_TOTAL 4096
#define N_BATCH 4
#define THREADS 256
#define DCHUNK  (THREADS * 4)     // 1024 channels per block
#define TCHUNK  128               // timesteps per block
#define PD      7                 // async prefetch depth
#define NBUF    8                 // LDS ring buffers (PD+1, power of 2)
#define ROWBYTES (THREADS * 16)   // 4096 B per buffered row slice

__device__ __forceinline__ void async_row_load(uint32_t lds_addr, const float* gaddr) {
    // VDST = per-lane LDS byte offset, VADDR = per-lane 64-bit global address.
    // Tracked with ASYNCcnt; each lane moves 16 B (wave moves 512 B).
    unsigned long long ga = (unsigned long long)(uintptr_t)gaddr;
    asm volatile("global_load_async_to_lds_b128 %0, %1, off"
                 :: "v"(lds_addr), "v"(ga)
                 : "memory");
}

__global__ __launch_bounds__(THREADS) void conv1d_dw_async_kernel(
    const float* __restrict__ x,
    const float* __restrict__ w,
    const float* __restrict__ bias,
    float* __restrict__ out)
{
    __shared__ v4f sbuf[NBUF * THREADS];

    const int tid = threadIdx.x;

    // Block decomposition: [batch][t-chunk][d-chunk]
    int blk = blockIdx.x;
    const int ndc = D_TOTAL / DCHUNK;   // 4
    const int ntc = T_TOTAL / TCHUNK;   // 32
    const int dc = blk % ndc; blk /= ndc;
    const int tc = blk % ntc; blk /= ntc;
    const int bb = blk;                 // 0..3

    const int d0 = dc * DCHUNK + tid * 4;
    const int t0 = tc * TCHUNK;

    const float* xb = x   + (size_t)bb * T_TOTAL * D_TOTAL;
    float*       ob = out + (size_t)bb * T_TOTAL * D_TOTAL;
    const float* gsrc = xb + (size_t)t0 * D_TOTAL + d0;

    // Per-thread taps and bias (constant over t).
    const v4f w0 = *(const v4f*)(w + 0 * D_TOTAL + d0);
    const v4f w1 = *(const v4f*)(w + 1 * D_TOTAL + d0);
    const v4f w2 = *(const v4f*)(w + 2 * D_TOTAL + d0);
    const v4f w3 = *(const v4f*)(w + 3 * D_TOTAL + d0);
    const v4f bv = *(const v4f*)(bias + d0);

    // Sliding window: rows t0-3..t0-1 (zero-padded at the sequence start).
    v4f xm3 = (v4f)0.0f, xm2 = (v4f)0.0f, xm1 = (v4f)0.0f;
    if (t0 > 0) {
        xm3 = *(const v4f*)(xb + (size_t)(t0 - 3) * D_TOTAL + d0);
        xm2 = *(const v4f*)(xb + (size_t)(t0 - 2) * D_TOTAL + d0);
        xm1 = *(const v4f*)(xb + (size_t)(t0 - 1) * D_TOTAL + d0);
    }

    // Low 32 bits of the generic shared-memory address are the LDS byte
    // offset (ISA 10.2).
    const uint32_t lbase = (uint32_t)(uintptr_t)(&sbuf[tid]);

    // Prologue: fill the async pipeline with rows t0 .. t0+PD-1.
#pragma unroll
    for (int j = 0; j < PD; ++j) {
        async_row_load(lbase + (uint32_t)j * ROWBYTES, gsrc + (size_t)j * D_TOTAL);
    }

    for (int i = 0; i < TCHUNK; ++i) {
        // Oldest in-flight async load (row i) is complete when <= PD-1 remain.
        asm volatile("s_wait_asynccnt 6" ::: "memory");

        const v4f cur = sbuf[(i & (NBUF - 1)) * THREADS + tid];

        v4f o = bv + w0 * xm3 + w1 * xm2 + w2 * xm1 + w3 * cur;

        __builtin_nontemporal_store(
            o, (v4f*)(ob + (size_t)(t0 + i) * D_TOTAL + d0));

        xm3 = xm2; xm2 = xm1; xm1 = cur;

        // Refill the ring. Target buffer (i+PD)&7 == (i-1)&7 was consumed
        // last iteration (its ds_load completed before that FMA), so this
        // async overwrite is hazard-free without barriers.
        const int pf = i + PD;
        if (pf < TCHUNK) {
            async_row_load(lbase + (uint32_t)(pf & (NBUF - 1)) * ROWBYTES,
                           gsrc + (size_t)pf * D_TOTAL);
        }
    }
}

extern "C" void kernel_launch(void* const* d_in, const int* in_sizes, int n_in,
                              void* d_out, int out_size, void* d_ws, size_t ws_size,
                              hipStream_t stream) {
    (void)in_sizes; (void)n_in; (void)out_size; (void)d_ws; (void)ws_size;
    const float* x = (const float*)d_in[0];
    const float* w = (const float*)d_in[1];
    const float* b = (const float*)d_in[2];
    float* out = (float*)d_out;

    const int nblocks = N_BATCH * (T_TOTAL / TCHUNK) * (D_TOTAL / DCHUNK); // 512
    conv1d_dw_async_kernel<<<dim3(nblocks), dim3(THREADS), 0, stream>>>(x, w, b, out);
}